// LocalAttentionModel_73856257622187
// MI455X (gfx1250) — compile-verified
//
#include <hip/hip_runtime.h>
#include <math.h>

// ---------------------------------------------------------------------------
// LocalAttentionModel for MI455X (gfx1250, wave32, WMMA)
//   qkv = mask(x) @ w_in^T + b_in        -> fp32 WMMA 16x16x4 GEMM
//   banded softmax-attention (win=3)     -> VALU (7/128 density; WMMA wasteful)
//   out = LN( attn_o @ w_out^T + b_out ) -> fp32 WMMA GEMM + fused LayerNorm
// fp32 WMMA chosen: problem is ~bandwidth bound (~1.7GB HBM @ 23.3TB/s),
// so we keep exact fp32 parity instead of chasing f16/fp8 compute peaks.
// ---------------------------------------------------------------------------

typedef float v2f __attribute__((ext_vector_type(2)));
typedef float v8f __attribute__((ext_vector_type(8)));

#define DIMD   320
#define SEQL   128
#define BSZ    8
#define HEADS  10
#define HD     32
#define WIN    3
#define MTOT   (BSZ * SEQL * SEQL)   // 131072 flattened rows
#define NQKV   (3 * DIMD)            // 960

// ===========================================================================
// Kernel 1: QKV GEMM.  C[M,960] = masked_x[M,320] @ w_in^T + b_in
// 256 threads = 8 waves: 4 row-groups x 2 col-groups. Block tile 64 x 192.
// K staged in LDS in chunks of 40 (10 WMMA k-steps per chunk).
// ===========================================================================
__global__ __launch_bounds__(256)
void qkv_gemm(const float* __restrict__ x, const unsigned char* __restrict__ pad,
              const float* __restrict__ w_in, const float* __restrict__ b_in,
              float* __restrict__ qkv)
{
    __shared__ float As[64][44];    // 44-float rows: 176B (16B aligned), bank-safe
    __shared__ float Ws[192][44];

    const int tid     = threadIdx.x;
    const int lane    = tid & 31;
    const int wave    = tid >> 5;
    const int waveRow = (wave & 3) << 4;    // 0,16,32,48
    const int waveCol = (wave >> 2) * 96;   // 0 or 96
    const int rowBase = blockIdx.x * 64;
    const int colBase = blockIdx.y * 192;
    const int l16     = lane & 15;
    const int koff    = (lane >> 4) << 1;   // lanes 16..31 take k+2,k+3
    const int rOff    = (lane >> 4) << 3;   // D rows 0..7 vs 8..15

    v8f acc[6];
    #pragma unroll
    for (int s = 0; s < 6; ++s) {
        float bv = b_in[colBase + waveCol + s * 16 + l16];
        #pragma unroll
        for (int e = 0; e < 8; ++e) acc[s][e] = bv;   // bias folded into C
    }

    for (int k0 = 0; k0 < DIMD; k0 += 40) {
        __syncthreads();
        // stage A tile 64x40 (float4), applying pad mask per row
        for (int i = tid; i < 640; i += 256) {
            int r = i / 10, c = (i % 10) * 4;
            int row = rowBase + r;
            float4 v = *(const float4*)&x[(size_t)row * DIMD + k0 + c];
            if (pad[row]) { v.x = 0.f; v.y = 0.f; v.z = 0.f; v.w = 0.f; }
            *(float4*)&As[r][c] = v;
        }
        // stage W tile 192x40 (w_in is [960,320] row-major = B^T, perfect layout)
        for (int i = tid; i < 1920; i += 256) {
            int r = i / 10, c = (i % 10) * 4;
            *(float4*)&Ws[r][c] = *(const float4*)&w_in[(size_t)(colBase + r) * DIMD + k0 + c];
        }
        __syncthreads();

        #pragma unroll
        for (int kk = 0; kk < 40; kk += 4) {
            v2f a = *(const v2f*)&As[waveRow + l16][kk + koff];
            #pragma unroll
            for (int s = 0; s < 6; ++s) {
                v2f b = *(const v2f*)&Ws[waveCol + s * 16 + l16][kk + koff];
                acc[s] = __builtin_amdgcn_wmma_f32_16x16x4_f32(
                    false, a, false, b, (short)0, acc[s], false, false);
            }
        }
    }

    #pragma unroll
    for (int s = 0; s < 6; ++s) {
        int col = colBase + waveCol + s * 16 + l16;
        #pragma unroll
        for (int v = 0; v < 8; ++v) {
            int row = rowBase + waveRow + rOff + v;
            qkv[(size_t)row * NQKV + col] = acc[s][v];
        }
    }
}

// ===========================================================================
// Kernel 2: banded local attention, window +/-3, softmax over <=7 keys.
// One block per (sequence n, head h); 128 threads = one query each.
// ===========================================================================
__global__ __launch_bounds__(128)
void local_attn(const float* __restrict__ qkv, float* __restrict__ o_mid)
{
    __shared__ float qs[SEQL][33];
    __shared__ float ks[SEQL][33];
    __shared__ float vs[SEQL][33];

    const int n   = blockIdx.x / HEADS;
    const int h   = blockIdx.x % HEADS;
    const int tid = threadIdx.x;
    const size_t base = (size_t)n * SEQL * NQKV + h * HD;

    for (int idx = tid; idx < SEQL * HD; idx += 128) {
        int l = idx >> 5, d = idx & 31;
        size_t g = base + (size_t)l * NQKV + d;
        qs[l][d] = qkv[g];
        ks[l][d] = qkv[g + DIMD];
        vs[l][d] = qkv[g + 2 * DIMD];
    }
    __syncthreads();

    const int i = tid;
    float q[HD];
    #pragma unroll
    for (int d = 0; d < HD; ++d) q[d] = qs[i][d];

    const float scale = 0.17677669529663687f;   // 1/sqrt(32)
    float sc[2 * WIN + 1];
    float mx = -3.0e38f;
    #pragma unroll
    for (int t = 0; t < 2 * WIN + 1; ++t) {
        int j = i - WIN + t;
        bool ok = (j >= 0) && (j < SEQL);
        int jc = ok ? j : i;
        float s = 0.f;
        #pragma unroll
        for (int d = 0; d < HD; ++d) s += q[d] * ks[jc][d];
        s = ok ? s * scale : -3.0e38f;
        sc[t] = s;
        mx = s > mx ? s : mx;
    }
    float sum = 0.f;
    #pragma unroll
    for (int t = 0; t < 2 * WIN + 1; ++t) {
        float e = expf(sc[t] - mx);   // masked taps underflow to exactly 0
        sc[t] = e;
        sum += e;
    }
    float inv = 1.f / sum;

    float o[HD];
    #pragma unroll
    for (int d = 0; d < HD; ++d) o[d] = 0.f;
    #pragma unroll
    for (int t = 0; t < 2 * WIN + 1; ++t) {
        int j = i - WIN + t;
        int jc = (j >= 0 && j < SEQL) ? j : i;   // weight is 0 when clamped
        float p = sc[t] * inv;
        #pragma unroll
        for (int d = 0; d < HD; ++d) o[d] += p * vs[jc][d];
    }

    float* dst = &o_mid[((size_t)n * SEQL + i) * DIMD + h * HD];
    #pragma unroll
    for (int d = 0; d < HD; ++d) dst[d] = o[d];
}

// ===========================================================================
// Kernel 3: out = LN( o_mid @ w_out^T + b_out ) * ln_g + ln_b
// 320 threads = 10 waves; block owns 16 full rows (all 320 cols) so the
// LayerNorm reduction is block-local. Wave w -> cols [32w, 32w+32).
// ===========================================================================
__global__ __launch_bounds__(320)
void out_gemm_ln(const float* __restrict__ a, const float* __restrict__ w_out,
                 const float* __restrict__ b_out, const float* __restrict__ ln_g,
                 const float* __restrict__ ln_b, float* __restrict__ out)
{
    __shared__ float As[16][44];
    __shared__ float Ws[320][44];
    __shared__ float Cs[16][DIMD];
    __shared__ float redS[16][20];
    __shared__ float redQ[16][20];
    __shared__ float muS[16];
    __shared__ float rsS[16];

    const int tid     = threadIdx.x;
    const int lane    = tid & 31;
    const int wave    = tid >> 5;           // 0..9
    const int l16     = lane & 15;
    const int koff    = (lane >> 4) << 1;
    const int rOff    = (lane >> 4) << 3;
    const int rowBase = blockIdx.x * 16;

    v8f acc[2];
    #pragma unroll
    for (int s = 0; s < 2; ++s) {
        float bv = b_out[wave * 32 + s * 16 + l16];
        #pragma unroll
        for (int e = 0; e < 8; ++e) acc[s][e] = bv;
    }

    for (int k0 = 0; k0 < DIMD; k0 += 40) {
        __syncthreads();
        if (tid < 160) {
            int r = tid / 10, c = (tid % 10) * 4;
            *(float4*)&As[r][c] = *(const float4*)&a[(size_t)(rowBase + r) * DIMD + k0 + c];
        }
        for (int i = tid; i < 3200; i += 320) {
            int r = i / 10, c = (i % 10) * 4;
            *(float4*)&Ws[r][c] = *(const float4*)&w_out[(size_t)r * DIMD + k0 + c];
        }
        __syncthreads();

        #pragma unroll
        for (int kk = 0; kk < 40; kk += 4) {
            v2f av = *(const v2f*)&As[l16][kk + koff];
            #pragma unroll
            for (int s = 0; s < 2; ++s) {
                v2f bv = *(const v2f*)&Ws[wave * 32 + s * 16 + l16][kk + koff];
                acc[s] = __builtin_amdgcn_wmma_f32_16x16x4_f32(
                    false, av, false, bv, (short)0, acc[s], false, false);
            }
        }
    }

    // park the 16x320 tile in LDS for the LayerNorm reduction
    #pragma unroll
    for (int s = 0; s < 2; ++s) {
        int col = wave * 32 + s * 16 + l16;
        #pragma unroll
        for (int v = 0; v < 8; ++v) Cs[rOff + v][col] = acc[s][v];
    }
    __syncthreads();

    // per-row mean/var: 20 threads per row, each sums 16 strided cols
    const int r   = tid / 20;
    const int t20 = tid % 20;
    float s1 = 0.f, s2 = 0.f;
    #pragma unroll
    for (int k = 0; k < 16; ++k) {
        float v = Cs[r][t20 + 20 * k];
        s1 += v;
        s2 += v * v;
    }
    redS[r][t20] = s1;
    redQ[r][t20] = s2;
    __syncthreads();
    if (t20 == 0) {
        float S = 0.f, Q = 0.f;
        #pragma unroll
        for (int q2 = 0; q2 < 20; ++q2) { S += redS[r][q2]; Q += redQ[r][q2]; }
        float mu  = S * (1.0f / DIMD);
        float var = Q * (1.0f / DIMD) - mu * mu;
        muS[r] = mu;
        rsS[r] = rsqrtf(var + 1e-5f);
    }
    __syncthreads();

    // normalize + affine; thread t owns column t for all 16 rows (coalesced)
    float g  = ln_g[tid];
    float bb = ln_b[tid];
    #pragma unroll
    for (int rr = 0; rr < 16; ++rr) {
        float v = (Cs[rr][tid] - muS[rr]) * rsS[rr] * g + bb;
        out[(size_t)(rowBase + rr) * DIMD + tid] = v;
    }
}

// ===========================================================================
extern "C" void kernel_launch(void* const* d_in, const int* in_sizes, int n_in,
                              void* d_out, int out_size, void* d_ws, size_t ws_size,
                              hipStream_t stream)
{
    const float*         x     = (const float*)d_in[0];
    const unsigned char* pad   = (const unsigned char*)d_in[1];  // bool mask
    const float*         w_in  = (const float*)d_in[2];
    const float*         b_in  = (const float*)d_in[3];
    const float*         w_out = (const float*)d_in[4];
    const float*         b_out = (const float*)d_in[5];
    const float*         ln_g  = (const float*)d_in[6];
    const float*         ln_b  = (const float*)d_in[7];
    float*               out   = (float*)d_out;

    float* qkv   = (float*)d_ws;                        // MTOT*960 f32 (503MB)
    float* o_mid = qkv + (size_t)MTOT * NQKV;           // MTOT*320 f32 (168MB)

    dim3 g1(MTOT / 64, NQKV / 192);                     // 2048 x 5
    qkv_gemm<<<g1, 256, 0, stream>>>(x, pad, w_in, b_in, qkv);

    local_attn<<<BSZ * SEQL * HEADS, 128, 0, stream>>>(qkv, o_mid);

    out_gemm_ln<<<MTOT / 16, 320, 0, stream>>>(o_mid, w_out, b_out, ln_g, ln_b, out);
}